// CAD_GCN_73882027426332
// MI455X (gfx1250) — compile-verified
//
#include <hip/hip_runtime.h>
#include <math.h>

typedef __attribute__((ext_vector_type(2))) float v2f;
typedef __attribute__((ext_vector_type(8))) float v8f;

constexpr int kB  = 16;
constexpr int kC  = 64;
constexpr int kHW = 256 * 256;   // 65536

// ---------------------------------------------------------------------------
// Stage A: global average pool. One block per (b,c) plane of 65536 floats.
// Pure streaming reduction: float4 loads, LDS tree reduce.
// ---------------------------------------------------------------------------
__global__ __launch_bounds__(256) void gcn_mean_kernel(const float* __restrict__ x,
                                                       float* __restrict__ xnode)
{
    __shared__ float red[256];
    const int plane = blockIdx.x;                       // b*64 + c
    const float4* p = (const float4*)(x + (size_t)plane * kHW);
    float s = 0.f;
#pragma unroll 4
    for (int i = 0; i < 64; ++i) {                      // 64 iters * 256 thr * 4 = 65536
        float4 v = p[i * 256 + threadIdx.x];
        s += (v.x + v.y) + (v.z + v.w);
    }
    red[threadIdx.x] = s;
    __syncthreads();
    for (int off = 128; off > 0; off >>= 1) {
        if (threadIdx.x < off) red[threadIdx.x] += red[threadIdx.x + off];
        __syncthreads();
    }
    if (threadIdx.x == 0) xnode[plane] = red[0] * (1.0f / (float)kHW);
}

// ---------------------------------------------------------------------------
// Stage B: node-space math on one wave32 using V_WMMA_F32_16X16X4_F32.
//   h     = x_node @ g1_w^T + g1_b        [16,64]
//   z1    = h * g2w + g2b                 [16,64]
//   w_eff = z1 @ theta_w                  [16,64]
//   bias_eff[b] = z1[b,:] . theta_b       [16]
// Fragment layouts per CDNA5 ISA 7.12.2:
//   A 16x4 (f32): lane&15 = M;  comp{0,1} hold K = k0 + 2*(lane>>4) + {0,1}
//   B 4x16 (f32): lane&15 = N;  comp{0,1} hold K = k0 + 2*(lane>>4) + {0,1}
//   C/D 16x16:    lane&15 = N;  vgpr v holds M = v + 8*(lane>>4)
// ---------------------------------------------------------------------------
__global__ __launch_bounds__(32) void gcn_graph_kernel(
    const float* __restrict__ xnode,    // [16][64]
    const float* __restrict__ theta_w,  // [64][64]
    const float* __restrict__ theta_b,  // [64]
    const float* __restrict__ g1_w,     // [64][64]
    const float* __restrict__ g1_b,     // [64]
    const float* __restrict__ g2_w,     // [1]
    const float* __restrict__ g2_b,     // [1]
    float* __restrict__ w_eff,          // [16][64]
    float* __restrict__ bias_eff)       // [16]
{
    __shared__ float xa[16 * 64];
    __shared__ float z [16 * 64];

    const int lane  = threadIdx.x;
    const int row   = lane & 15;          // M for A, N for B and C/D
    const int khalf = (lane >> 4) * 2;    // K sub-offset within a 4-wide K step
    const int mhi   = (lane >> 4) * 8;    // M offset for C/D rows

    for (int i = lane; i < 16 * 64; i += 32) xa[i] = xnode[i];
    __syncthreads();

    const float g2w = g2_w[0];
    const float g2b = g2_b[0];

    // ---- h / z1 : x_node @ g1_w^T ----
    for (int nt = 0; nt < 4; ++nt) {
        v8f acc = {};
#pragma unroll
        for (int k0 = 0; k0 < 64; k0 += 4) {
            v2f a, b;
            a.x = xa[row * 64 + k0 + khalf];
            a.y = xa[row * 64 + k0 + khalf + 1];
            // B[k][n] = g1_w[nt*16+n][k]  (h = x_node @ g1_w^T)
            b.x = g1_w[(nt * 16 + row) * 64 + k0 + khalf];
            b.y = g1_w[(nt * 16 + row) * 64 + k0 + khalf + 1];
            acc = __builtin_amdgcn_wmma_f32_16x16x4_f32(false, a, false, b,
                                                        (short)0, acc, false, false);
        }
        const int   col  = nt * 16 + row;
        const float bias = g1_b[col];
#pragma unroll
        for (int v = 0; v < 8; ++v) {
            float hv = acc[v] + bias;
            z[(v + mhi) * 64 + col] = hv * g2w + g2b;   // z1
        }
    }
    __syncthreads();

    // ---- w_eff : z1 @ theta_w ----
    for (int ct = 0; ct < 4; ++ct) {
        v8f acc = {};
#pragma unroll
        for (int k0 = 0; k0 < 64; k0 += 4) {
            v2f a, b;
            a.x = z[row * 64 + k0 + khalf];
            a.y = z[row * 64 + k0 + khalf + 1];
            // B[k][c] = theta_w[k][ct*16 + c]
            b.x = theta_w[(k0 + khalf)     * 64 + ct * 16 + row];
            b.y = theta_w[(k0 + khalf + 1) * 64 + ct * 16 + row];
            acc = __builtin_amdgcn_wmma_f32_16x16x4_f32(false, a, false, b,
                                                        (short)0, acc, false, false);
        }
#pragma unroll
        for (int v = 0; v < 8; ++v)
            w_eff[(v + mhi) * 64 + ct * 16 + row] = acc[v];
    }

    // ---- bias_eff[b] = z1[b,:] . theta_b ----  (after all WMMAs; EXEC may diverge now)
    if (lane < 16) {
        float s = 0.f;
        for (int n = 0; n < 64; ++n) s += z[lane * 64 + n] * theta_b[n];
        bias_eff[lane] = s;
    }
}

// ---------------------------------------------------------------------------
// Stage C: streaming epilogue. One thread per pixel; 64 channel values cached
// in registers so x is read exactly once here. Coalesced along W.
//   y   = w_eff[b] . x[b,:,h,w] + bias_eff[b]
//   out = tanh(x + y*phi_w[c] + phi_b[c])
// ---------------------------------------------------------------------------
__global__ __launch_bounds__(256) void gcn_apply_kernel(
    const float* __restrict__ x,
    const float* __restrict__ w_eff,
    const float* __restrict__ bias_eff,
    const float* __restrict__ phi_w,
    const float* __restrict__ phi_b,
    float* __restrict__ out)
{
    __shared__ float sw[64], spw[64], spb[64];
    const int b   = blockIdx.x >> 8;      // 256 blocks per batch image
    const int blk = blockIdx.x & 255;
    if (threadIdx.x < 64) {
        sw [threadIdx.x] = w_eff[b * 64 + threadIdx.x];
        spw[threadIdx.x] = phi_w[threadIdx.x];
        spb[threadIdx.x] = phi_b[threadIdx.x];
    }
    __syncthreads();

    const int    hw   = blk * 256 + threadIdx.x;            // 0..65535
    const size_t base = (size_t)b * 64 * (size_t)kHW + hw;

    float v[64];
#pragma unroll
    for (int c = 0; c < 64; ++c)
        v[c] = x[base + (size_t)c * kHW];

    float y = bias_eff[b];
#pragma unroll
    for (int c = 0; c < 64; ++c)
        y += sw[c] * v[c];

#pragma unroll
    for (int c = 0; c < 64; ++c)
        out[base + (size_t)c * kHW] = tanhf(v[c] + y * spw[c] + spb[c]);
}

// ---------------------------------------------------------------------------
extern "C" void kernel_launch(void* const* d_in, const int* in_sizes, int n_in,
                              void* d_out, int out_size, void* d_ws, size_t ws_size,
                              hipStream_t stream)
{
    const float* x       = (const float*)d_in[0];
    const float* theta_w = (const float*)d_in[1];
    const float* theta_b = (const float*)d_in[2];
    const float* g1_w    = (const float*)d_in[3];
    const float* g1_b    = (const float*)d_in[4];
    const float* g2_w    = (const float*)d_in[5];
    const float* g2_b    = (const float*)d_in[6];
    const float* phi_w   = (const float*)d_in[7];
    const float* phi_b   = (const float*)d_in[8];
    float* out = (float*)d_out;

    float* ws       = (float*)d_ws;
    float* xnode    = ws;          // 16*64 floats
    float* w_eff    = ws + 1024;   // 16*64 floats
    float* bias_eff = ws + 2048;   // 16 floats

    gcn_mean_kernel <<<kB * kC, 256, 0, stream>>>(x, xnode);
    gcn_graph_kernel<<<1, 32, 0, stream>>>(xnode, theta_w, theta_b,
                                           g1_w, g1_b, g2_w, g2_b,
                                           w_eff, bias_eff);
    gcn_apply_kernel<<<kB * 256, 256, 0, stream>>>(x, w_eff, bias_eff,
                                                   phi_w, phi_b, out);
}